// pRNN_37452114821540
// MI455X (gfx1250) — compile-verified
//
#include <hip/hip_runtime.h>

// pRNN: 1024 sequential steps, each: gather (256,32) from past step outputs,
// relu((256,32) @ W[i]^T + b[i]) -> trace[i]; return trace[1023].
//
// Single-WGP persistent kernel, latency-optimized:
//  - 2-deep LDS ring forwards the last two steps' outputs (no L2 round trip
//    on the tight dependence chain).
//  - global gathers only touch src <= i-2; their producers' ASYNC stores are
//    guaranteed complete by a relaxed s_wait_asynccnt 8 one step earlier.
//  - trace[i] is pushed to global by GLOBAL_STORE_ASYNC_FROM_LDS_B128 (TDM-
//    class async path, ASYNCcnt) straight from the LDS ring: 8 x 16B per lane
//    instead of 32 scalar stores, each wave DMAing only its own row block.
//  - next step's conn / gather / W-fragments / bias prefetched during the
//    current step's WMMA phase (W lives in registers, never in LDS).
//  - raw s_barrier + manual counter waits (no compiler-forced full fences).

#define NN 1024
#define BB 256
#define DD 64
#define KK 32
#define THREADS 512
#define NWAVE (THREADS / 32)          // 16 waves: one 16-row M-tile each
#define GITS ((BB * KK) / THREADS)    // 16 gather iterations per lane
#define BD (BB * DD)

typedef __attribute__((ext_vector_type(2))) float v2f;
typedef __attribute__((ext_vector_type(8))) float v8f;

static __device__ __forceinline__ float agent_load(const float* p) {
    return __hip_atomic_load(p, __ATOMIC_RELAXED, __HIP_MEMORY_SCOPE_AGENT);
}

// LDS(ring block) -> global, async DMA path, device-scope completion.
static __device__ __forceinline__ void async_store_b128(const void* gaddr,
                                                        unsigned lds_byte_addr) {
    asm volatile("global_store_async_from_lds_b128 %0, %1, off scope:SCOPE_DEV"
                 :: "v"((unsigned long long)(uintptr_t)gaddr), "v"(lds_byte_addr)
                 : "memory");
}

__launch_bounds__(THREADS, 1)
__global__ void prnn_kernel(const float* __restrict__ x,
                            const float* __restrict__ W,
                            const float* __restrict__ b,
                            const int*   __restrict__ conn_src,
                            const int*   __restrict__ conn_k,
                            float*       __restrict__ trace,   // d_ws: (N-1)*B*D floats
                            float*       __restrict__ out)     // d_out: B*D floats
{
    extern __shared__ float smem[];
    float* lds_vals = smem;            // BB*KK floats (32 KB): gathered A operand
    float* lds_ring = smem + BB * KK;  // 2*BD floats (128 KB): out[i], out[i-1]

    const int tid    = threadIdx.x;
    const int wave   = tid >> 5;
    const int lane   = tid & 31;
    const int l15    = lane & 15;
    const int hi     = lane >> 4;            // A/B fragment half: K=0,1 vs K=2,3
    const int a_row  = wave * 16 + l15;      // wave == M-tile
    const int a_koff = hi * 2;
    const int m_base = wave * 16 + hi * 8;   // C/D: VGPR r -> row m_base + r
    int b_col[4];
    #pragma unroll
    for (int n = 0; n < 4; ++n) b_col[n] = n * 16 + l15;

    // ---------------- prologue: prefetch step-0 operands ----------------
    float vold[GITS];                        // step 0 gathers straight from x
    #pragma unroll
    for (int it = 0; it < GITS; ++it)
        vold[it] = x[(wave + it * NWAVE) * KK + lane];

    v2f wf[4][8];                            // B fragments of W[i] (registers only)
    #pragma unroll
    for (int n = 0; n < 4; ++n)
        #pragma unroll
        for (int kc = 0; kc < 8; ++kc)
            wf[n][kc] = *(const v2f*)&W[b_col[n] * KK + kc * 4 + a_koff];

    float bias[4];
    #pragma unroll
    for (int n = 0; n < 4; ++n) bias[n] = b[b_col[n]];

    int cls = 2, rsel = 0, ckk = 0;          // step 0: all lanes take vold (x)

    for (int i = 0; i < NN; ++i) {
        // ---- (A) finalize gather into lds_vals: {0 | LDS ring | prefetched} ----
        #pragma unroll
        for (int it = 0; it < GITS; ++it) {
            const int bidx = wave + it * NWAVE;
            float v = 0.0f;
            if (cls == 2)      v = vold[it];                              // src <= i-3 (or x)
            else if (cls == 1) v = lds_ring[rsel * BD + bidx * DD + ckk]; // src in {i-1,i-2}
            lds_vals[bidx * KK + lane] = v;                               // src >= i -> 0
        }
        asm volatile("s_wait_dscnt 0" ::: "memory");
        __builtin_amdgcn_s_barrier();

        // ---- (B1) prefetch next step's gather (old sources only) ----
        int nsrc = 0, nkk = 0;
        if (i + 1 < NN) {
            nsrc = conn_src[(i + 1) * KK + lane];
            nkk  = conn_k [(i + 1) * KK + lane];
            if (nsrc <= i - 2) {   // async stores of step nsrc completed by end of step i-1
                const float* gp = trace + (long)nsrc * BD + nkk;
                #pragma unroll
                for (int it = 0; it < GITS; ++it)
                    vold[it] = agent_load(gp + (wave + it * NWAVE) * DD);
            }
        }

        // ---- (B2) GEMM: (256x32)@(32x64) via V_WMMA_F32_16X16X4_F32 ----
        v8f c[4];
        #pragma unroll
        for (int n = 0; n < 4; ++n) {
            const float bv = bias[n];
            c[n] = (v8f){bv, bv, bv, bv, bv, bv, bv, bv};
        }
        #pragma unroll
        for (int kc = 0; kc < 8; ++kc) {
            const v2f a = *(const v2f*)&lds_vals[a_row * KK + kc * 4 + a_koff];
            #pragma unroll
            for (int n = 0; n < 4; ++n)
                c[n] = __builtin_amdgcn_wmma_f32_16x16x4_f32(
                           false, a, false, wf[n][kc], (short)0, c[n], false, false);
        }

        // ---- (B3) ReLU + store to LDS ring only (global copy goes async) ----
        float* ldst = lds_ring + (i & 1) * BD;
        #pragma unroll
        for (int r = 0; r < 8; ++r) {
            const int row = m_base + r;
            #pragma unroll
            for (int n = 0; n < 4; ++n)
                ldst[row * DD + b_col[n]] = fmaxf(c[n][r], 0.0f);
        }

        // ---- (B4) prefetch next W fragments / bias; classify next gather ----
        if (i + 1 < NN) {
            const float* wp = W + (long)(i + 1) * (DD * KK);
            #pragma unroll
            for (int n = 0; n < 4; ++n)
                #pragma unroll
                for (int kc = 0; kc < 8; ++kc)
                    wf[n][kc] = *(const v2f*)&wp[b_col[n] * KK + kc * 4 + a_koff];
            #pragma unroll
            for (int n = 0; n < 4; ++n) bias[n] = b[(i + 1) * DD + b_col[n]];
            cls  = (nsrc >= i + 1) ? 0 : ((nsrc >= i - 1) ? 1 : 2);
            rsel = nsrc & 1;
            ckk  = nkk;
        }

        // ---- (B5) async DMA this wave's own 16x64 ring block to global ----
        // own-wave ds writes committed first, so no cross-wave LDS hazard.
        asm volatile("s_wait_dscnt 0" ::: "memory");
        {
            float* gdst = (i == NN - 1) ? out : (trace + (long)i * BD);
            const unsigned blk = (unsigned)(wave * 16 * DD * sizeof(float)); // 4 KB/wave
            const unsigned lds_base = __builtin_amdgcn_groupstaticsize()
                                    + (unsigned)((BB * KK + (i & 1) * BD) * sizeof(float))
                                    + blk;
            const char* gbase = (const char*)gdst + blk;
            #pragma unroll
            for (int c2 = 0; c2 < 8; ++c2) {
                const unsigned off = (unsigned)((c2 * 32 + lane) * 16);
                async_store_b128(gbase + off, lds_base + off);
            }
        }

        // PREVIOUS step's 8 async stores complete (this step's may stay in
        // flight -> one-step-lagged global visibility, matching src <= i-2);
        // barrier also publishes the ring for the next step's gather.
        asm volatile("s_wait_asynccnt 8" ::: "memory");
        __builtin_amdgcn_s_barrier();
    }
}

extern "C" void kernel_launch(void* const* d_in, const int* in_sizes, int n_in,
                              void* d_out, int out_size, void* d_ws, size_t ws_size,
                              hipStream_t stream) {
    (void)in_sizes; (void)n_in; (void)out_size; (void)ws_size;
    const float* x        = (const float*)d_in[0];
    const float* W        = (const float*)d_in[1];
    const float* b        = (const float*)d_in[2];
    const int*   conn_src = (const int*)  d_in[3];
    const int*   conn_k   = (const int*)  d_in[4];
    float*       trace    = (float*)d_ws;   // (N-1)*B*D*4 ~= 64 MB scratch (L2-resident)
    float*       out      = (float*)d_out;

    const size_t lds_bytes = (size_t)(BB * KK + 2 * BD) * sizeof(float); // 160 KB of 320 KB WGP LDS
    (void)hipFuncSetAttribute((const void*)prnn_kernel,
                              hipFuncAttributeMaxDynamicSharedMemorySize,
                              (int)lds_bytes);
    prnn_kernel<<<1, THREADS, lds_bytes, stream>>>(x, W, b, conn_src, conn_k, trace, out);
}